// LocalAttention_4784593567863
// MI455X (gfx1250) — compile-verified
//
#include <hip/hip_runtime.h>

// ---------------------------------------------------------------------------
// Local attention (bug-faithful bucketed reshape) for MI455X / gfx1250.
// All matmuls run on V_WMMA_F32_16X16X4_F32 (exact fp32 matrix pipe).
// ---------------------------------------------------------------------------

typedef float v2f __attribute__((ext_vector_type(2)));
typedef float v8f __attribute__((ext_vector_type(8)));

#define NEG_MAX (-3.402823466e38f)

__device__ __forceinline__ v8f wmma4(v2f a, v2f b, v8f c) {
  // D = A(16x4) * B(4x16) + C(16x16), fp32, wave32.
  return __builtin_amdgcn_wmma_f32_16x16x4_f32(
      /*neg_a=*/false, a, /*neg_b=*/false, b,
      /*c_mod=*/(short)0, c, /*reuse_a=*/false, /*reuse_b=*/false);
}

__device__ __forceinline__ v8f vzero8() {
  v8f z;
#pragma unroll
  for (int q = 0; q < 8; ++q) z[q] = 0.f;
  return z;
}

// LDS tile row stride (floats): multiple of 4 (16B-aligned float4 stores),
// 36r mod 64 distinct for r=0..15 -> conflict-free b64 fragment reads.
#define LDW 36

// ---------------------------------------------------------------------------
// Phase 1: qkv = x @ w_qkv^T, scatter-stored directly into bucketed layouts.
//   Qb/Kb: [bw=B*32+w][i(128)][e(1024)]      (row-major, e contiguous)
//   Vtb  : [bw][e(1024)][i(128)]             (transposed for the AV GEMM)
// Index map (faithful to the reference reshape):
//   w = 2h + (t>>11), i = (t>>4)&127, e = ((t&15)<<6) + d
// ---------------------------------------------------------------------------
__global__ __launch_bounds__(256) void gemm_qkv_kernel(
    const float* __restrict__ X, const float* __restrict__ W,
    float* __restrict__ Qb, float* __restrict__ Kb, float* __restrict__ Vtb) {
  __shared__ float As[128 * LDW];
  __shared__ float Bs[128 * LDW];
  const int K = 1024;
  const int m0 = blockIdx.y * 128;
  const int n0 = blockIdx.x * 128;
  const int tid = threadIdx.x;
  const int wave = tid >> 5;
  const int lane = tid & 31;
  const int lrow = lane & 15;
  const int coff0 = (lane >> 4) * 2;
  const int wm0 = (wave >> 2) * 64;
  const int wn0 = (wave & 3) * 32;

  v8f acc[4][2];
#pragma unroll
  for (int a = 0; a < 4; ++a)
#pragma unroll
    for (int b = 0; b < 2; ++b) acc[a][b] = vzero8();

  for (int k0 = 0; k0 < K; k0 += 32) {
#pragma unroll
    for (int it = 0; it < 4; ++it) {
      int f = tid + it * 256;
      int r = f >> 3;
      int c = (f & 7) << 2;
      *(float4*)&As[r * LDW + c] =
          *(const float4*)&X[(size_t)(m0 + r) * K + k0 + c];
      *(float4*)&Bs[r * LDW + c] =
          *(const float4*)&W[(size_t)(n0 + r) * K + k0 + c];
    }
    if (k0 + 32 < K) {  // emits global_prefetch_b8 on gfx1250
      __builtin_prefetch(&X[(size_t)(m0 + (tid >> 1)) * K + k0 + 32], 0, 0);
      __builtin_prefetch(&W[(size_t)(n0 + (tid >> 1)) * K + k0 + 32], 0, 0);
    }
    __syncthreads();
#pragma unroll
    for (int kk = 0; kk < 32; kk += 4) {
      v2f af[4], bf[2];
#pragma unroll
      for (int mt = 0; mt < 4; ++mt)
        af[mt] = *(const v2f*)&As[(wm0 + mt * 16 + lrow) * LDW + kk + coff0];
#pragma unroll
      for (int nt = 0; nt < 2; ++nt)
        bf[nt] = *(const v2f*)&Bs[(wn0 + nt * 16 + lrow) * LDW + kk + coff0];
#pragma unroll
      for (int mt = 0; mt < 4; ++mt)
#pragma unroll
        for (int nt = 0; nt < 2; ++nt)
          acc[mt][nt] = wmma4(af[mt], bf[nt], acc[mt][nt]);
    }
    __syncthreads();
  }

  const int mhigh = lane >> 4;
  const int nloc = lane & 15;
#pragma unroll
  for (int mt = 0; mt < 4; ++mt) {
#pragma unroll
    for (int nt = 0; nt < 2; ++nt) {
      const int c = n0 + wn0 + nt * 16 + nloc;
      const int sel = c >> 10;   // 0=q, 1=k, 2=v  (uniform per tile)
      const int cc = c & 1023;
      const int h = cc >> 6;
      const int d = cc & 63;
#pragma unroll
      for (int r = 0; r < 8; ++r) {
        const int m = m0 + wm0 + mt * 16 + r + 8 * mhigh;
        const int Bbat = m >> 12;
        const int t = m & 4095;
        const int w = 2 * h + (t >> 11);
        const int i = (t >> 4) & 127;
        const int e = ((t & 15) << 6) + d;
        const size_t bw = (size_t)(Bbat * 32 + w);
        const float val = acc[mt][nt][r];
        if (sel == 0)      Qb[(bw * 128 + i) * 1024 + e] = val;
        else if (sel == 1) Kb[(bw * 128 + i) * 1024 + e] = val;
        else               Vtb[(bw * 1024 + e) * 128 + i] = val;
      }
    }
  }
}

// ---------------------------------------------------------------------------
// Phase 2a: energy = scale * Q @ K_la^T, masks, softmax -> attn (global).
// One block per (B,w). Each wave owns 16 query rows x all 256 key columns,
// so the softmax reductions are wave-local (half-wave shfl_xor, width 16).
// ---------------------------------------------------------------------------
__global__ __launch_bounds__(256) void attn_energy_kernel(
    const float* __restrict__ Qb, const float* __restrict__ Kb,
    const unsigned char* __restrict__ mask, float* __restrict__ attn) {
  __shared__ float Qs[128 * LDW];
  __shared__ float Ks[256 * LDW];
  const int bw = blockIdx.x;     // B*32 + w
  const int Bbat = bw >> 5;
  const int w = bw & 31;
  const int tid = threadIdx.x;
  const int wave = tid >> 5;
  const int lane = tid & 31;
  const int lrow = lane & 15;
  const int coff0 = (lane >> 4) * 2;
  const bool haveprev = (w > 0);

  const float* qbase = Qb + (size_t)bw * 128 * 1024;
  const float* kcur  = Kb + (size_t)bw * 128 * 1024;
  const float* kprev = Kb + (size_t)(bw - 1) * 128 * 1024;  // used only if w>0

  v8f acc[16];
#pragma unroll
  for (int t = 0; t < 16; ++t) acc[t] = vzero8();

  for (int e0 = 0; e0 < 1024; e0 += 32) {
#pragma unroll
    for (int it = 0; it < 4; ++it) {
      int f = tid + it * 256;
      int r = f >> 3;
      int c = (f & 7) << 2;
      *(float4*)&Qs[r * LDW + c] =
          *(const float4*)&qbase[(size_t)r * 1024 + e0 + c];
    }
#pragma unroll
    for (int it = 0; it < 8; ++it) {
      int f = tid + it * 256;
      int r = f >> 3;          // 0..255 (look-around row)
      int c = (f & 7) << 2;
      float4 v;
      if (r < 128) {           // previous window (pad_value = -1.0 at w==0)
        if (haveprev) v = *(const float4*)&kprev[(size_t)r * 1024 + e0 + c];
        else          v = make_float4(-1.f, -1.f, -1.f, -1.f);
      } else {                 // current window
        v = *(const float4*)&kcur[(size_t)(r - 128) * 1024 + e0 + c];
      }
      *(float4*)&Ks[r * LDW + c] = v;
    }
    __syncthreads();
#pragma unroll
    for (int kk = 0; kk < 32; kk += 4) {
      v2f a = *(const v2f*)&Qs[(wave * 16 + lrow) * LDW + kk + coff0];
#pragma unroll
      for (int t = 0; t < 16; ++t) {
        v2f b = *(const v2f*)&Ks[(t * 16 + lrow) * LDW + kk + coff0];
        acc[t] = wmma4(a, b, acc[t]);
      }
    }
    __syncthreads();
  }

  // --- scale + masks (natural positions) ---
  const float scale = 0.03125f;  // 1024^-0.5
  const int mhigh = lane >> 4;
  const int nloc = lane & 15;
  const unsigned char* mrow = mask + (size_t)Bbat * 4096;

  bool qok[8];
#pragma unroll
  for (int r = 0; r < 8; ++r) {
    const int i = wave * 16 + r + 8 * mhigh;
    qok[r] = (mrow[w * 128 + i] != 0);
  }

#pragma unroll
  for (int t = 0; t < 16; ++t) {
    const int j = t * 16 + nloc;
    bool kok;
    if (j < 128) kok = haveprev && (mrow[(w - 1) * 128 + j] != 0);
    else         kok = (mrow[w * 128 + (j - 128)] != 0);
#pragma unroll
    for (int r = 0; r < 8; ++r) {
      const int i = wave * 16 + r + 8 * mhigh;
      const bool bad = (!kok) || (!qok[r]) || ((j >= 128) && ((j - 128) > i));
      acc[t][r] = bad ? NEG_MAX : acc[t][r] * scale;
    }
  }

  // --- softmax over j (wave-local: 16 tiles x half-wave lanes) ---
  float pmax[8], psum[8];
#pragma unroll
  for (int r = 0; r < 8; ++r) pmax[r] = NEG_MAX;
#pragma unroll
  for (int t = 0; t < 16; ++t)
#pragma unroll
    for (int r = 0; r < 8; ++r) pmax[r] = fmaxf(pmax[r], acc[t][r]);
#pragma unroll
  for (int off = 8; off >= 1; off >>= 1)
#pragma unroll
    for (int r = 0; r < 8; ++r)
      pmax[r] = fmaxf(pmax[r], __shfl_xor(pmax[r], off, 16));

#pragma unroll
  for (int r = 0; r < 8; ++r) psum[r] = 0.f;
#pragma unroll
  for (int t = 0; t < 16; ++t)
#pragma unroll
    for (int r = 0; r < 8; ++r) {
      const float ex = __expf(acc[t][r] - pmax[r]);
      acc[t][r] = ex;
      psum[r] += ex;
    }
#pragma unroll
  for (int off = 8; off >= 1; off >>= 1)
#pragma unroll
    for (int r = 0; r < 8; ++r) psum[r] += __shfl_xor(psum[r], off, 16);

  float* arow = attn + (size_t)bw * 128 * 256;
#pragma unroll
  for (int r = 0; r < 8; ++r) {
    const int i = wave * 16 + r + 8 * mhigh;
    const float inv = 1.f / psum[r];
#pragma unroll
    for (int t = 0; t < 16; ++t) {
      const int j = t * 16 + nloc;
      arow[(size_t)i * 256 + j] = acc[t][r] * inv;
    }
  }
}

// ---------------------------------------------------------------------------
// Phase 2b: out_w = attn @ V_la, stored directly into merged (B,t,h*64+d).
// Grid: x = e-chunk (4 x 256), y = bw. NT GEMM via pre-transposed Vtb.
// ---------------------------------------------------------------------------
__global__ __launch_bounds__(256) void attn_av_kernel(
    const float* __restrict__ attn, const float* __restrict__ Vtb,
    float* __restrict__ om) {
  __shared__ float As[128 * LDW];
  __shared__ float Bs[256 * LDW];
  const int bw = blockIdx.y;
  const int Bbat = bw >> 5;
  const int w = bw & 31;
  const int e0 = blockIdx.x * 256;
  const int tid = threadIdx.x;
  const int wave = tid >> 5;
  const int lane = tid & 31;
  const int lrow = lane & 15;
  const int coff0 = (lane >> 4) * 2;
  const int wm0 = (wave >> 2) * 64;  // i sub-block
  const int wn0 = (wave & 3) * 64;   // e sub-block
  const bool haveprev = (w > 0);

  const float* ag = attn + (size_t)bw * 128 * 256;
  const float* vcur = Vtb + (size_t)bw * 1024 * 128;
  const float* vprev = Vtb + (size_t)(bw - 1) * 1024 * 128;  // only if w>0

  v8f acc[4][4];
#pragma unroll
  for (int a = 0; a < 4; ++a)
#pragma unroll
    for (int b = 0; b < 4; ++b) acc[a][b] = vzero8();

  for (int j0 = 0; j0 < 256; j0 += 32) {
#pragma unroll
    for (int it = 0; it < 4; ++it) {
      int f = tid + it * 256;
      int r = f >> 3;
      int c = (f & 7) << 2;
      *(float4*)&As[r * LDW + c] =
          *(const float4*)&ag[(size_t)r * 256 + j0 + c];
    }
#pragma unroll
    for (int it = 0; it < 8; ++it) {
      int f = tid + it * 256;
      int r = f >> 3;          // e-local 0..255
      int c = (f & 7) << 2;
      const int e = e0 + r;
      float4 v;
      if (j0 < 128) {          // look-back half (attn there is 0 when w==0)
        if (haveprev) v = *(const float4*)&vprev[(size_t)e * 128 + j0 + c];
        else          v = make_float4(-1.f, -1.f, -1.f, -1.f);
      } else {
        v = *(const float4*)&vcur[(size_t)e * 128 + (j0 - 128) + c];
      }
      *(float4*)&Bs[r * LDW + c] = v;
    }
    __syncthreads();
#pragma unroll
    for (int kk = 0; kk < 32; kk += 4) {
      v2f af[4], bf[4];
#pragma unroll
      for (int mt = 0; mt < 4; ++mt)
        af[mt] = *(const v2f*)&As[(wm0 + mt * 16 + lrow) * LDW + kk + coff0];
#pragma unroll
      for (int nt = 0; nt < 4; ++nt)
        bf[nt] = *(const v2f*)&Bs[(wn0 + nt * 16 + lrow) * LDW + kk + coff0];
#pragma unroll
      for (int mt = 0; mt < 4; ++mt)
#pragma unroll
        for (int nt = 0; nt < 4; ++nt)
          acc[mt][nt] = wmma4(af[mt], bf[nt], acc[mt][nt]);
    }
    __syncthreads();
  }

  // store into merged layout: out_merged[B, t, h*64+d]
  const int mhigh = lane >> 4;
  const int nloc = lane & 15;
  const int h = w >> 1;
  const int tbase = (w & 1) << 11;
#pragma unroll
  for (int mt = 0; mt < 4; ++mt) {
#pragma unroll
    for (int nt = 0; nt < 4; ++nt) {
      const int ecol = e0 + wn0 + nt * 16 + nloc;
      const int d = ecol & 63;
      const int ehi = ecol >> 6;
#pragma unroll
      for (int r = 0; r < 8; ++r) {
        const int i = wm0 + mt * 16 + r + 8 * mhigh;
        const int t = tbase + i * 16 + ehi;
        om[(size_t)(Bbat * 4096 + t) * 1024 + h * 64 + d] = acc[mt][nt][r];
      }
    }
  }
}

// ---------------------------------------------------------------------------
// Phase 3: final = out_merged @ w_fc^T + b_fc
// ---------------------------------------------------------------------------
__global__ __launch_bounds__(256) void gemm_fc_kernel(
    const float* __restrict__ A, const float* __restrict__ W,
    const float* __restrict__ bias, float* __restrict__ Out) {
  __shared__ float As[128 * LDW];
  __shared__ float Bs[128 * LDW];
  const int K = 1024;
  const int m0 = blockIdx.y * 128;
  const int n0 = blockIdx.x * 128;
  const int tid = threadIdx.x;
  const int wave = tid >> 5;
  const int lane = tid & 31;
  const int lrow = lane & 15;
  const int coff0 = (lane >> 4) * 2;
  const int wm0 = (wave >> 2) * 64;
  const int wn0 = (wave & 3) * 32;

  v8f acc[4][2];
#pragma unroll
  for (int a = 0; a < 4; ++a)
#pragma unroll
    for (int b = 0; b < 2; ++b) acc[a][b] = vzero8();

  for (int k0 = 0; k0 < K; k0 += 32) {
#pragma unroll
    for (int it = 0; it < 4; ++it) {
      int f = tid + it * 256;
      int r = f >> 3;
      int c = (f & 7) << 2;
      *(float4*)&As[r * LDW + c] =
          *(const float4*)&A[(size_t)(m0 + r) * K + k0 + c];
      *(float4*)&Bs[r * LDW + c] =
          *(const float4*)&W[(size_t)(n0 + r) * K + k0 + c];
    }
    if (k0 + 32 < K) {
      __builtin_prefetch(&A[(size_t)(m0 + (tid >> 1)) * K + k0 + 32], 0, 0);
      __builtin_prefetch(&W[(size_t)(n0 + (tid >> 1)) * K + k0 + 32], 0, 0);
    }
    __syncthreads();
#pragma unroll
    for (int kk = 0; kk < 32; kk += 4) {
      v2f af[4], bf[2];
#pragma unroll
      for (int mt = 0; mt < 4; ++mt)
        af[mt] = *(const v2f*)&As[(wm0 + mt * 16 + lrow) * LDW + kk + coff0];
#pragma unroll
      for (int nt = 0; nt < 2; ++nt)
        bf[nt] = *(const v2f*)&Bs[(wn0 + nt * 16 + lrow) * LDW + kk + coff0];
#pragma unroll
      for (int mt = 0; mt < 4; ++mt)
#pragma unroll
        for (int nt = 0; nt < 2; ++nt)
          acc[mt][nt] = wmma4(af[mt], bf[nt], acc[mt][nt]);
    }
    __syncthreads();
  }

  const int mhigh = lane >> 4;
  const int nloc = lane & 15;
#pragma unroll
  for (int mt = 0; mt < 4; ++mt) {
#pragma unroll
    for (int nt = 0; nt < 2; ++nt) {
      const int c = n0 + wn0 + nt * 16 + nloc;
      const float bv = bias[c];
#pragma unroll
      for (int r = 0; r < 8; ++r) {
        const int m = m0 + wm0 + mt * 16 + r + 8 * mhigh;
        Out[(size_t)m * 1024 + c] = acc[mt][nt][r] + bv;
      }
    }
  }
}

// ---------------------------------------------------------------------------
extern "C" void kernel_launch(void* const* d_in, const int* in_sizes, int n_in,
                              void* d_out, int out_size, void* d_ws,
                              size_t ws_size, hipStream_t stream) {
  (void)in_sizes; (void)n_in; (void)out_size; (void)ws_size;

  const float* x = (const float*)d_in[0];               // [8,4096,1024]
  const unsigned char* mask = (const unsigned char*)d_in[1];  // [8,4096] bool
  const float* w_qkv = (const float*)d_in[2];           // [3072,1024]
  const float* w_fc = (const float*)d_in[3];            // [1024,1024]
  const float* b_fc = (const float*)d_in[4];            // [1024]

  float* out = (float*)d_out;                           // 33,554,432 f32
  float* attn = out + (size_t)8 * 4096 * 1024;          //  8,388,608 f32

  const size_t SEG = (size_t)33554432;                  // 128 MB each
  float* Qb = (float*)d_ws;
  float* Kb = Qb + SEG;
  float* Vtb = Kb + SEG;
  float* om = Vtb + SEG;

  // 1) QKV projection, scatter into bucketed layouts
  gemm_qkv_kernel<<<dim3(24, 256), 256, 0, stream>>>(x, w_qkv, Qb, Kb, Vtb);
  // 2a) energy + masks + softmax -> attn output
  attn_energy_kernel<<<dim3(256), 256, 0, stream>>>(Qb, Kb, mask, attn);
  // 2b) attn @ V -> merged (B,t,channel) layout
  attn_av_kernel<<<dim3(4, 256), 256, 0, stream>>>(attn, Vtb, om);
  // 3) output projection + bias
  gemm_fc_kernel<<<dim3(8, 256), 256, 0, stream>>>(om, w_fc, b_fc, out);
}